// LipschitzNorm_28819230556489
// MI455X (gfx1250) — compile-verified
//
#include <hip/hip_runtime.h>
#include <stdint.h>

#define HEADS 4
#define DFEAT 32
#define NNODES 50000
#define EPSV 1e-12f

// Padded row stride in LDS: 32 floats + 4 floats pad = 36 dwords (144 B).
// gcd(36,64)=4 -> b128 reads across a wave hit 16 distinct 4-bank groups,
// worst case 2-way conflict.
#define ROW_DW 36

// ---------------------------------------------------------------------------
// Phase 1: async global->LDS stream of x (CDNA5 ASYNCcnt path), thread-per-row
// squared-norm from padded LDS, coalesced norm_x store + scatter atomic max.
// ---------------------------------------------------------------------------
__global__ __launch_bounds__(256) void lips_rownorm_scatter_max(
    const float* __restrict__ x, const int* __restrict__ index,
    float* __restrict__ norm_x, unsigned int* __restrict__ maxn, int R)
{
  __shared__ float tile[256 * ROW_DW];        // 36 KB
  const int t = threadIdx.x;
  const int tileIdx = blockIdx.x;
  const int tileChunkBase = tileIdx * 2048;   // 16 B chunks per tile
  const int totalChunks = R * 8;              // R rows * 128 B / 16 B
  const unsigned ldsBase = (unsigned)(uintptr_t)&tile[0];

  // Async fill: chunk c = (row<<3)|j of the tile; global source is row-major,
  // LDS destination uses the padded row stride. Issue order is coalesced:
  // iteration i covers 256 consecutive 16 B chunks (512 B per wave op).
  if (tileChunkBase + 2048 <= totalChunks) {  // full tile: no guards
    #pragma unroll
    for (int i = 0; i < 8; ++i) {
      const int c = i * 256 + t;
      const unsigned lds_off =
          ldsBase + (unsigned)(c >> 3) * (ROW_DW * 4u) + (unsigned)(c & 7) * 16u;
      const int gbyte = (tileChunkBase + c) * 16;   // < 2^31
      asm volatile("global_load_async_to_lds_b128 %0, %1, %2"
                   :: "v"(lds_off), "v"(gbyte), "s"(x) : "memory");
    }
  } else {                                     // tail tile: guarded
    #pragma unroll
    for (int i = 0; i < 8; ++i) {
      const int c = i * 256 + t;
      if (tileChunkBase + c < totalChunks) {
        const unsigned lds_off =
            ldsBase + (unsigned)(c >> 3) * (ROW_DW * 4u) + (unsigned)(c & 7) * 16u;
        const int gbyte = (tileChunkBase + c) * 16;
        asm volatile("global_load_async_to_lds_b128 %0, %1, %2"
                     :: "v"(lds_off), "v"(gbyte), "s"(x) : "memory");
      }
    }
  }
  asm volatile("s_wait_asynccnt 0x0" ::: "memory");
  __syncthreads();

  // Thread t owns local row t: 8x ds_load_b128 + 32 FMAs (4 accumulators).
  const float4* rowp = (const float4*)&tile[t * ROW_DW];
  float a0 = 0.f, a1 = 0.f, a2 = 0.f, a3 = 0.f;
  #pragma unroll
  for (int j = 0; j < 8; ++j) {
    const float4 v = rowp[j];
    a0 = fmaf(v.x, v.x, a0);
    a1 = fmaf(v.y, v.y, a1);
    a2 = fmaf(v.z, v.z, a2);
    a3 = fmaf(v.w, v.w, a3);
  }
  const float acc = (a0 + a1) + (a2 + a3);

  const int r = tileIdx * 256 + t;             // coalesced
  if (r < R) {
    norm_x[r] = acc;
    const int node = index[r >> 2];            // H == 4
    // norms >= 0 and table zero-initialized: u32 order == f32 order;
    // max is order-independent -> deterministic across replays.
    atomicMax(&maxn[node * HEADS + (r & 3)], __float_as_uint(acc));
  }
}

// ---------------------------------------------------------------------------
// Phase 2: norm_att[h] = 4 * || concat(att_l[h,:], att_r[h,:]) ||   (tiny)
// ---------------------------------------------------------------------------
__global__ __launch_bounds__(128) void lips_att_norm(
    const float* __restrict__ att_l, const float* __restrict__ att_r,
    float* __restrict__ att_s)
{
  const int t = threadIdx.x;                   // 4 waves, one per head
  const int h = t >> 5, d = t & 31;
  const float a = att_l[h * DFEAT + d];
  const float b = att_r[h * DFEAT + d];
  float s = a * a + b * b;
  #pragma unroll
  for (int o = 16; o > 0; o >>= 1) s += __shfl_xor(s, o, 32);
  if (d == 0) att_s[h] = 4.0f * sqrtf(s);
}

// ---------------------------------------------------------------------------
// Phase 3: out[e,h] = alpha[e,h] / (att_s[h]*sqrt(maxn[index[e]] + norm_x) + eps)
// (bug-faithful flat gather: index[e] indexes the flattened [N,H] table)
// ---------------------------------------------------------------------------
__global__ __launch_bounds__(256) void lips_finalize(
    const float* __restrict__ alpha, const int* __restrict__ index,
    const float* __restrict__ norm_x, const float* __restrict__ maxn,
    const float* __restrict__ att_s, float* __restrict__ out, int R)
{
  const int r = blockIdx.x * 256 + threadIdx.x;
  if (r >= R) return;
  const float nx = norm_x[r];
  const float m  = maxn[index[r >> 2]];        // 800 KB table -> L2 resident
  const float denom = att_s[r & 3] * sqrtf(m + nx) + EPSV;
  out[r] = alpha[r] / denom;
}

// ---------------------------------------------------------------------------
extern "C" void kernel_launch(void* const* d_in, const int* in_sizes, int n_in,
                              void* d_out, int out_size, void* d_ws, size_t ws_size,
                              hipStream_t stream) {
  const float* x     = (const float*)d_in[0];   // [E,H,D]
  const float* att_l = (const float*)d_in[1];   // [1,H,D]
  const float* att_r = (const float*)d_in[2];   // [1,H,D]
  const float* alpha = (const float*)d_in[3];   // [E,H]
  const int*   index = (const int*)d_in[4];     // [E]
  const int E = in_sizes[4];
  const int R = E * HEADS;                      // total (e,h) rows

  // Workspace layout: [maxn: N*H u32][norm_x: R f32][att_s: H f32]
  unsigned char* ws = (unsigned char*)d_ws;
  const size_t maxnBytes = (size_t)NNODES * HEADS * sizeof(unsigned int);
  unsigned int* maxn   = (unsigned int*)ws;
  float*        norm_x = (float*)(ws + maxnBytes);
  float*        att_s  = (float*)(ws + maxnBytes + (size_t)R * sizeof(float));

  hipMemsetAsync(maxn, 0, maxnBytes, stream);   // 0.0f == segment_max empty->0

  const int tiles = (R + 255) / 256;            // 256 rows (32 KB of x) per block
  lips_rownorm_scatter_max<<<tiles, 256, 0, stream>>>(x, index, norm_x, maxn, R);
  lips_att_norm<<<1, 128, 0, stream>>>(att_l, att_r, att_s);
  lips_finalize<<<(R + 255) / 256, 256, 0, stream>>>(
      alpha, index, norm_x, (const float*)maxn, att_s, (float*)d_out, R);
}